// String_Finder_69741678953239
// MI455X (gfx1250) — compile-verified
//
#include <hip/hip_runtime.h>
#include <hip/hip_bf16.h>

#define EPSF 1e-6f
#define SQRT2F 1.41421356237f

__device__ __forceinline__ float safef(float x) {
    return (fabsf(x) < EPSF) ? EPSF : x;
}

__global__ void __launch_bounds__(256)
string_finder_kernel(const float* __restrict__ strs,
                     const int*   __restrict__ edges,
                     float* __restrict__ out_new,   // (E,7)
                     float* __restrict__ out_spl,   // (E,10,2)
                     float* __restrict__ out_sim,   // (E,)
                     int nE, int nS)
{
    int e = blockIdx.x * blockDim.x + threadIdx.x;
    if (e >= nE) return;

    int i0 = edges[2 * e + 0];
    int i1 = edges[2 * e + 1];
    bool cond_lf = (i0 >= nS);
    bool cond_rt = (i1 >= nS);
    int j0 = cond_lf ? (i0 - nS) : i0;
    int j1 = cond_rt ? (i1 - nS) : i1;

    const float* __restrict__ p0 = strs + (size_t)j0 * 7;
    const float* __restrict__ p1 = strs + (size_t)j1 * 7;
    // gfx1250 global_prefetch_b8: get both gathered cachelines in flight together
    __builtin_prefetch(p0, 0, 1);
    __builtin_prefetch(p1, 0, 1);

    float s_lf[7], s_rt[7];
#pragma unroll
    for (int k = 0; k < 7; ++k) { s_lf[k] = p0[k]; s_rt[k] = p1[k]; }

    // close_* = all_ends[i] == (cond ? row[2:4] : row[0:2]) of the SAME gathered row
    float clx = cond_lf ? s_lf[2] : s_lf[0];
    float cly = cond_lf ? s_lf[3] : s_lf[1];
    float crx = cond_rt ? s_rt[2] : s_rt[0];
    float cry = cond_rt ? s_rt[3] : s_rt[1];
    // NOTE: crosswise conditions, exactly as in the reference
    float flx = cond_rt ? s_lf[2] : s_lf[0];
    float fly = cond_rt ? s_lf[3] : s_lf[1];
    float frx = cond_lf ? s_rt[2] : s_rt[0];
    float fry = cond_lf ? s_rt[3] : s_rt[1];

    float svx = flx - frx + 1e-6f;
    float svy = fly - fry + 1e-6f;
    float seg_len = sqrtf(svx * svx + svy * svy);
    bool  not_coinc = seg_len > 0.1f;
    float safe_len = not_coinc ? seg_len : 1.0f;

    float d0x = (clx + crx) * 0.5f;
    float d0y = (cly + cry) * 0.5f;
    float dev_dists = (fry - fly) * (flx - d0x) - (fly - d0y) * (frx - flx);
    float dev_frac = dev_dists / safe_len;

    float snx = (s_lf[4] + s_rt[4]) * 0.5f;
    float sny = (s_lf[5] + s_rt[5]) * 0.5f;
    float nlen = safef(sqrtf(snx * snx + sny * sny));
    snx /= nlen; sny /= nlen;

    float vx = frx - flx, vy = fry - fly;
    float px = vy / safe_len;    // exact div: feeds the sqrt2 sign decision
    float py = -vx / safe_len;
    float ndx = snx - px, ndy = sny - py;
    float norm_diff = sqrtf(ndx * ndx + ndy * ndy);
    float nnx, nny;
    if (norm_diff < SQRT2F) { nnx = px;  nny = py;  }
    else                    { nnx = -px; nny = -py; }

    float seg_cx = (flx + frx) * 0.5f;
    float seg_cy = (fly + fry) * 0.5f;
    float amp = seg_len * dev_frac;
    float dlx = seg_cx + nnx * amp;
    float dly = seg_cy + nny * amp;

    float c = nnx, s = nny;
    // rot @ v = (c*vx + s*vy, -s*vx + c*vy)
    float rlfx =  c * flx + s * fly, rlfy = -s * flx + c * fly;
    float rrtx =  c * frx + s * fry, rrty = -s * frx + c * fry;
    float rdvx =  c * dlx + s * dly, rdvy = -s * dlx + c * dly;

    float ddx = rrtx - rlfx, ddy = rrty - rlfy;
    float dlen = sqrtf(ddx * ddx + ddy * ddy);

    // stable 3-element sort keyed on locs_x=[rlf.y, rdev.y, rrt.y], payload locs_y
    float x0 = rlfy, x1 = rdvy, x2 = rrty;
    float y0 = rlfx, y1 = rdvx, y2 = rrtx;
    float tx, ty;
    if (x1 < x0) { tx = x0; x0 = x1; x1 = tx; ty = y0; y0 = y1; y1 = ty; }
    if (x2 < x1) {
        tx = x1; x1 = x2; x2 = tx; ty = y1; y1 = y2; y2 = ty;
        if (x1 < x0) { tx = x0; x0 = x1; x1 = tx; ty = y0; y0 = y1; y1 = ty; }
    }
    float x_least = x0, y_least = y0;
    float xs1 = x1 - x_least, xs2 = x2 - x_least;
    float ys1 = y1 - y_least, ys2 = y2 - y_least;

    // Hoist the Lagrange denominators: 2 precise divides instead of 20
    float r1 = 1.0f / (safef(xs1) * safef(xs1 - xs2));
    float r2 = 1.0f / (safef(xs2) * safef(xs2 - xs1));

    // sim + mask
    float llx = s_lf[0] - s_lf[2] + 1e-6f, lly = s_lf[1] - s_lf[3] + 1e-6f;
    float len_l = sqrtf(llx * llx + lly * lly);
    float lrx = s_rt[0] - s_rt[2] + 1e-6f, lry = s_rt[1] - s_rt[3] + 1e-6f;
    float len_r = sqrtf(lrx * lrx + lry * lry);
    float n2x = s_lf[4] - s_rt[4] + 1e-6f, n2y = s_lf[5] - s_rt[5] + 1e-6f;
    float norm_dist = sqrtf(n2x * n2x + n2y * n2y);
    float sim = s_lf[6] / (s_rt[6] + 0.001f) + len_l / (len_r + 0.001f) + norm_dist;
    bool  m = not_coinc && (sim > 1.0f);

    // new_strs output (E,7), masked
    float* __restrict__ on = out_new + (size_t)e * 7;
    on[0] = m ? flx : 0.0f;
    on[1] = m ? fly : 0.0f;
    on[2] = m ? frx : 0.0f;
    on[3] = m ? fry : 0.0f;
    on[4] = m ? nnx : 0.0f;
    on[5] = m ? nny : 0.0f;
    on[6] = m ? dev_frac : 0.0f;

    // splines output (E,10,2), masked
    float* __restrict__ os = out_spl + (size_t)e * (2 * 10);
#pragma unroll
    for (int k = 0; k < 10; ++k) {
        float tt = ((float)k / 9.0f) * dlen;
        float L1 = tt * (tt - xs2) * r1;
        float L2 = tt * (tt - xs1) * r2;
        float iy = ys1 * L1 + ys2 * L2;
        float spx = tt + x_least;
        float spy = iy + y_least;
        float o0 =  c * spx + s * spy;   // splines[...,0] = rot row0 . spl
        float o1 = -s * spx + c * spy;   // splines[...,1] = rot row1 . spl
        os[2 * k + 0] = m ? o0 : 0.0f;
        os[2 * k + 1] = m ? o1 : 0.0f;
    }

    out_sim[e] = m ? sim : 0.0f;
}

extern "C" void kernel_launch(void* const* d_in, const int* in_sizes, int n_in,
                              void* d_out, int out_size, void* d_ws, size_t ws_size,
                              hipStream_t stream) {
    const float* strs  = (const float*)d_in[0];   // (N_STRS, 7) float32
    const int*   edges = (const int*)d_in[1];     // (N_EDGES, 2) int32

    int nS = in_sizes[0] / 7;   // 500000
    int nE = in_sizes[1] / 2;   // 1000000

    float* out = (float*)d_out;
    float* out_new = out;                         // nE*7
    float* out_spl = out + (size_t)nE * 7;        // nE*20
    float* out_sim = out + (size_t)nE * 27;       // nE

    const int BLK = 256;
    int grid = (nE + BLK - 1) / BLK;
    string_finder_kernel<<<grid, BLK, 0, stream>>>(strs, edges,
                                                   out_new, out_spl, out_sim,
                                                   nE, nS);
}